// VectorQuantizerEMA_4896262717887
// MI455X (gfx1250) — compile-verified
//
#include <hip/hip_runtime.h>

typedef float v2f __attribute__((ext_vector_type(2)));
typedef float v8f __attribute__((ext_vector_type(8)));

#define NROWS   65536
#define DDIM    256
#define KCODES  1024
#define CHUNK   128          // embedding codes staged in LDS per chunk
#define PITCH   260          // padded row pitch (floats): 260 % 64 == 4 -> conflict-free ds_load_b64
#define ROWS_PER_WG 128      // 8 waves * 16 rows

// ---------------------------------------------------------------------------
// Kernel 0: per-code squared norms  |e_k|^2
// ---------------------------------------------------------------------------
__global__ void __launch_bounds__(256) vq_enorm_kernel(const float* __restrict__ emb,
                                                       float* __restrict__ eNorm) {
    int code = blockIdx.x * 256 + threadIdx.x;   // grid covers 1024 codes
    const float4* e4 = reinterpret_cast<const float4*>(emb + (size_t)code * DDIM);
    float s = 0.0f;
#pragma unroll
    for (int i = 0; i < DDIM / 4; ++i) {
        float4 v = e4[i];
        s += v.x * v.x + v.y * v.y + v.z * v.z + v.w * v.w;
    }
    eNorm[code] = s;
}

// ---------------------------------------------------------------------------
// Kernel 1: fused distance GEMM (f32 WMMA) + row-wise argmin
//   dist(m,k) = |e_k|^2 - 2 * x_m . e_k        (|x_m|^2 dropped: argmin-invariant)
// ---------------------------------------------------------------------------
__global__ void __launch_bounds__(256) vq_argmin_kernel(const float* __restrict__ x,
                                                        const float* __restrict__ emb,
                                                        const float* __restrict__ eNorm,
                                                        int* __restrict__ outIdx) {
    extern __shared__ float lds[];               // CHUNK * PITCH floats (130 KB)

    const int tid  = threadIdx.x;
    const int lane = tid & 31;
    const int wave = tid >> 5;                   // 0..7
    const int n    = lane & 15;                  // N column within 16-wide tile
    const int h    = lane >> 4;                  // half-wave select (K-pair {0,1} vs {2,3})
    const int rowBase = blockIdx.x * ROWS_PER_WG + wave * 16;
    const int myRow   = rowBase + n;             // A-matrix M = lane%16

    // --- preload this wave's 16x256 A strip: 64 fragments, ISA 32-bit 16x4 layout ---
    v2f a[64];
    {
        const float* xrow = x + (size_t)myRow * DDIM + 2 * h;
#pragma unroll
        for (int ks = 0; ks < 64; ++ks)
            a[ks] = *reinterpret_cast<const v2f*>(xrow + 4 * ks);
    }

    float minVal[8];
    int   minIdx[8];
#pragma unroll
    for (int v = 0; v < 8; ++v) { minVal[v] = 3.4e38f; minIdx[v] = 0; }

    for (int chunk = 0; chunk < KCODES / CHUNK; ++chunk) {
        const int codeBase = chunk * CHUNK;

        __syncthreads();
        // --- cooperative staging: 128 codes x 256 f32 -> LDS, pitch 260 ---
        {
            const float4* src = reinterpret_cast<const float4*>(emb + (size_t)codeBase * DDIM);
#pragma unroll
            for (int it = 0; it < (CHUNK * DDIM / 4) / 256; ++it) {   // 32 iters
                int slot = tid + 256 * it;                            // float4 slot
                int code = slot >> 6;                                 // 64 float4 per code
                int dpos = (slot & 63) << 2;
                float4 vv = src[(size_t)code * (DDIM / 4) + (slot & 63)];
                *reinterpret_cast<float4*>(&lds[code * PITCH + dpos]) = vv;   // ds_store_b128
            }
        }
        __syncthreads();

#pragma unroll 1
        for (int t = 0; t < CHUNK / 16; ++t) {
            v8f acc0 = {};
            v8f acc1 = {};
            // B fragment: lane reads e[16t+n][4ks + 2h .. +1]  (conflict-free ds_load_b64)
            const float* brow = &lds[(t * 16 + n) * PITCH + 2 * h];
#pragma unroll
            for (int ks = 0; ks < 64; ks += 2) {
                v2f b0 = *reinterpret_cast<const v2f*>(brow + 4 * ks);
                v2f b1 = *reinterpret_cast<const v2f*>(brow + 4 * (ks + 1));
                acc0 = __builtin_amdgcn_wmma_f32_16x16x4_f32(false, a[ks],     false, b0,
                                                             (short)0, acc0, false, false);
                acc1 = __builtin_amdgcn_wmma_f32_16x16x4_f32(false, a[ks + 1], false, b1,
                                                             (short)0, acc1, false, false);
            }
            const int   code = codeBase + t * 16 + n;
            const float en   = eNorm[code];
            v8f dot = acc0 + acc1;
#pragma unroll
            for (int v = 0; v < 8; ++v) {
                float dval = en - 2.0f * dot[v];
                // codes strictly increase per (lane,v): '<' keeps first-minimum
                if (dval < minVal[v]) { minVal[v] = dval; minIdx[v] = code; }
            }
        }
    }

    // --- argmin reduction over the 16 lanes sharing each M row (xor butterfly) ---
#pragma unroll
    for (int v = 0; v < 8; ++v) {
        float mv = minVal[v];
        int   mi = minIdx[v];
#pragma unroll
        for (int m = 1; m < 16; m <<= 1) {
            float ov = __shfl_xor(mv, m, 32);
            int   oi = __shfl_xor(mi, m, 32);
            if (ov < mv || (ov == mv && oi < mi)) { mv = ov; mi = oi; }
        }
        minVal[v] = mv;
        minIdx[v] = mi;
    }
    // D-tile layout: lanes 0-15 hold M=v, lanes 16-31 hold M=v+8
    if (n == 0) {
#pragma unroll
        for (int v = 0; v < 8; ++v)
            outIdx[rowBase + h * 8 + v] = minIdx[v];
    }
}

// ---------------------------------------------------------------------------
// Kernel 2: streaming outputs. Block = 256 threads = 4 rows x 64 lanes x float4.
// ---------------------------------------------------------------------------
__global__ void __launch_bounds__(256) vq_output_kernel(const float* __restrict__ x,
                                                        const float* __restrict__ emb,
                                                        const int* __restrict__ idx,
                                                        float* __restrict__ loss,
                                                        float* __restrict__ qst,
                                                        float* __restrict__ onehot) {
    const int tid = threadIdx.x;
    const int r   = tid >> 6;                    // row within block (0..3)
    const int c   = (tid & 63) << 2;             // float column (0..252)
    const int row = blockIdx.x * 4 + r;
    const int code = idx[row];

    const float4 xv = *reinterpret_cast<const float4*>(x   + (size_t)row  * DDIM + c);
    const float4 qv = *reinterpret_cast<const float4*>(emb + (size_t)code * DDIM + c);

    float4 dv, lv, sv;
    dv.x = qv.x - xv.x; dv.y = qv.y - xv.y; dv.z = qv.z - xv.z; dv.w = qv.w - xv.w;
    lv.x = 0.25f * dv.x * dv.x; lv.y = 0.25f * dv.y * dv.y;
    lv.z = 0.25f * dv.z * dv.z; lv.w = 0.25f * dv.w * dv.w;
    // match reference rounding: qst = x + (q - x)
    sv.x = xv.x + dv.x; sv.y = xv.y + dv.y; sv.z = xv.z + dv.z; sv.w = xv.w + dv.w;

    *reinterpret_cast<float4*>(loss + (size_t)row * DDIM + c) = lv;
    *reinterpret_cast<float4*>(qst  + (size_t)row * DDIM + c) = sv;

    // one-hot row: 1024 floats = 64 lanes * float4 * 4 iters
#pragma unroll
    for (int it = 0; it < 4; ++it) {
        int col = c + it * 256;
        float4 ov;
        ov.x = (col + 0 == code) ? 1.0f : 0.0f;
        ov.y = (col + 1 == code) ? 1.0f : 0.0f;
        ov.z = (col + 2 == code) ? 1.0f : 0.0f;
        ov.w = (col + 3 == code) ? 1.0f : 0.0f;
        *reinterpret_cast<float4*>(onehot + (size_t)row * KCODES + col) = ov;
    }
}

// ---------------------------------------------------------------------------
extern "C" void kernel_launch(void* const* d_in, const int* in_sizes, int n_in,
                              void* d_out, int out_size, void* d_ws, size_t ws_size,
                              hipStream_t stream) {
    (void)in_sizes; (void)n_in; (void)out_size; (void)ws_size;

    const float* x   = (const float*)d_in[0];   // (65536, 256)
    const float* emb = (const float*)d_in[1];   // (1024, 256)

    float* eNorm   = (float*)d_ws;                                   // 1024 f32
    int*   indices = (int*)((char*)d_ws + KCODES * sizeof(float));   // 65536 i32

    float* loss   = (float*)d_out;                                   // 65536*256
    float* qst    = loss + (size_t)NROWS * DDIM;                     // 65536*256
    float* onehot = qst  + (size_t)NROWS * DDIM;                     // 65536*1024

    vq_enorm_kernel<<<KCODES / 256, 256, 0, stream>>>(emb, eNorm);

    const size_t ldsBytes = (size_t)CHUNK * PITCH * sizeof(float);   // 133120 B
    vq_argmin_kernel<<<NROWS / ROWS_PER_WG, 256, ldsBytes, stream>>>(x, emb, eNorm, indices);

    vq_output_kernel<<<NROWS / 4, 256, 0, stream>>>(x, emb, indices, loss, qst, onehot);
}